// CoreferenceResolver_90099823935818
// MI455X (gfx1250) — compile-verified
//
#include <hip/hip_runtime.h>
#include <hip/hip_bf16.h>
#include <math.h>

typedef __attribute__((ext_vector_type(2))) float v2f;
typedef __attribute__((ext_vector_type(8))) float v8f;

constexpr int Bc = 8;
constexpr int Lc = 8192;
constexpr int Dc = 768;
constexpr int NEc = 128;
constexpr int NPAIR = 8128;            // NE*(NE-1)/2
constexpr int NTILE = 64;              // 8x8 tiles of 16x16 per batch

// workspace layout (in floats)
constexpr size_t EHAT_OFF = 0;                         // [B][NE][D] normalized embeddings
constexpr size_t EHAT_SZ  = (size_t)Bc * NEc * Dc;     // 786432
constexpr size_t SIM_OFF  = EHAT_OFF + EHAT_SZ;        // [B][NE][NE]
constexpr size_t SIM_SZ   = (size_t)Bc * NEc * NEc;    // 131072
constexpr size_t PART_OFF = SIM_OFF + SIM_SZ;          // [B][64][2] per-tile (sum, sumsq)
constexpr size_t PART_SZ  = (size_t)Bc * NTILE * 2;
constexpr size_t STD_OFF  = PART_OFF + PART_SZ;        // [B] std

// ---------------------------------------------------------------------------
// Kernel 1: span mean-pool + L2-normalize. One block per (b, entity).
// Gathers only the span rows (<=16 rows of 768 f32) instead of the
// reference's masked [NE,L]x[L,D] matmul over all of x.
// ---------------------------------------------------------------------------
__global__ void __launch_bounds__(256)
pool_norm_kernel(const float* __restrict__ x,
                 const int* __restrict__ starts,
                 const int* __restrict__ lengths,
                 float* __restrict__ ehat) {
    __shared__ float red[256];
    const int be  = blockIdx.x;          // b*NE + e
    const int b   = be >> 7;             // / NE
    const int s   = starts[be];
    const int len = lengths[be];
    const int tid = threadIdx.x;

    const float* base = x + ((size_t)b * Lc + s) * Dc;
    float a0 = 0.f, a1 = 0.f, a2 = 0.f;
    for (int r = 0; r < len; ++r) {
        const float* row = base + (size_t)r * Dc;
        a0 += row[tid];
        a1 += row[tid + 256];
        a2 += row[tid + 512];
    }
    const float inv = 1.0f / (float)len;
    a0 *= inv; a1 *= inv; a2 *= inv;

    red[tid] = a0 * a0 + a1 * a1 + a2 * a2;
    __syncthreads();
    for (int off = 128; off > 0; off >>= 1) {
        if (tid < off) red[tid] += red[tid + off];
        __syncthreads();
    }
    // sim denominator is max(n_i*n_j, 1e-8); normalizing rows by
    // rsqrt(max(|e|^2, 1e-16)) reproduces that to fp32 accuracy.
    const float rs = rsqrtf(fmaxf(red[0], 1e-16f));

    float* out = ehat + (size_t)be * Dc;
    out[tid]       = a0 * rs;
    out[tid + 256] = a1 * rs;
    out[tid + 512] = a2 * rs;
}

// ---------------------------------------------------------------------------
// Kernel 2: sim = Ehat * Ehat^T via V_WMMA_F32_16X16X4_F32 (exact fp32 MMA).
// One wave (32 lanes) per 16x16 output tile; K-loop of 192 WMMAs over D=768.
// Also emits deterministic per-tile (sum, sumsq) partials for the std.
// ---------------------------------------------------------------------------
__global__ void __launch_bounds__(32)
sim_wmma_kernel(const float* __restrict__ ehat,
                float* __restrict__ sim,
                float* __restrict__ part) {
    const int blk  = blockIdx.x;         // b*64 + tile
    const int b    = blk >> 6;
    const int t    = blk & 63;
    const int tm   = t >> 3;
    const int tn   = t & 7;
    const int lane = threadIdx.x;
    const int lrow  = lane & 15;
    const int khalf = (lane >> 4) * 2;   // fp32 A/B 16x4 layout: lanes 16-31 hold K+2,K+3

    const float* Arow = ehat + (size_t)(b * NEc + tm * 16 + lrow) * Dc + khalf;
    const float* Brow = ehat + (size_t)(b * NEc + tn * 16 + lrow) * Dc + khalf;

    v8f c = {};
#pragma unroll 8
    for (int k = 0; k < Dc; k += 4) {
        v2f a  = *(const v2f*)(Arow + k);
        v2f bb = *(const v2f*)(Brow + k);
        // 8 args: (neg_a, A, neg_b, B, c_mod, C, reuse_a, reuse_b)
        c = __builtin_amdgcn_wmma_f32_16x16x4_f32(false, a, false, bb,
                                                  (short)0, c, false, false);
    }

    // C layout: VGPR r, lanes 0-15 -> M = r, lanes 16-31 -> M = 8 + r; N = lane&15.
    const int col   = tn * 16 + lrow;
    const int rbase = tm * 16 + (lane >> 4) * 8;
    float s = 0.f, sq = 0.f;
#pragma unroll
    for (int r = 0; r < 8; ++r) {
        float v = c[r];
        sim[((size_t)b * NEc + (rbase + r)) * NEc + col] = v;
        s  += v;
        sq += v * v;
    }
#pragma unroll
    for (int m = 16; m > 0; m >>= 1) {
        s  += __shfl_xor(s,  m, 32);
        sq += __shfl_xor(sq, m, 32);
    }
    if (lane == 0) {
        part[(size_t)(b * NTILE + t) * 2 + 0] = s;
        part[(size_t)(b * NTILE + t) * 2 + 1] = sq;
    }
}

// ---------------------------------------------------------------------------
// Kernel 3: per-batch unbiased std over the 128x128 sim matrix from tile
// partials (deterministic tree reduction).
// ---------------------------------------------------------------------------
__global__ void __launch_bounds__(64)
std_kernel(const float* __restrict__ part, float* __restrict__ stdArr) {
    __shared__ float ss[64], qq[64];
    const int b = blockIdx.x;
    const int tid = threadIdx.x;
    ss[tid] = part[(size_t)(b * NTILE + tid) * 2 + 0];
    qq[tid] = part[(size_t)(b * NTILE + tid) * 2 + 1];
    __syncthreads();
    for (int off = 32; off > 0; off >>= 1) {
        if (tid < off) { ss[tid] += ss[tid + off]; qq[tid] += qq[tid + off]; }
        __syncthreads();
    }
    if (tid == 0) {
        const float n = (float)(NEc * NEc);
        const float mean = ss[0] / n;
        const float var  = (qq[0] - n * mean * mean) / (n - 1.0f);
        stdArr[b] = sqrtf(fmaxf(var, 0.f));
    }
}

// ---------------------------------------------------------------------------
// Kernel 4: gather i<j pairs, standardize, rank-1 MLP (1->128 ReLU ->2).
// One thread per pair; weights are uniform -> scalar cache loads.
// ---------------------------------------------------------------------------
__global__ void __launch_bounds__(256)
pair_mlp_kernel(const float* __restrict__ sim,
                const float* __restrict__ stdArr,
                const int* __restrict__ hts,
                const float* __restrict__ thr,
                const float* __restrict__ W1,
                const float* __restrict__ b1,
                const float* __restrict__ W2,
                const float* __restrict__ b2,
                float* __restrict__ out) {
    const int idx = blockIdx.x * blockDim.x + threadIdx.x;
    const int total = Bc * NPAIR;
    if (idx >= total) return;
    const int b = idx / NPAIR;
    const int p = idx - b * NPAIR;
    const int i = hts[((size_t)b * NPAIR + p) * 2 + 0];
    const int j = hts[((size_t)b * NPAIR + p) * 2 + 1];

    const float sv  = sim[((size_t)b * NEc + i) * NEc + j];
    const float val = (sv - thr[0]) / (stdArr[b] + 1e-5f);

    float l0 = b2[0], l1 = b2[1];
#pragma unroll 4
    for (int k = 0; k < 128; ++k) {
        float h = fmaxf(fmaf(val, W1[k], b1[k]), 0.f);
        l0 = fmaf(h, W2[2 * k + 0], l0);
        l1 = fmaf(h, W2[2 * k + 1], l1);
    }
    out[2 * idx + 0] = l0;
    out[2 * idx + 1] = l1;
}

// ---------------------------------------------------------------------------
extern "C" void kernel_launch(void* const* d_in, const int* in_sizes, int n_in,
                              void* d_out, int out_size, void* d_ws, size_t ws_size,
                              hipStream_t stream) {
    const float* x       = (const float*)d_in[0];
    const int*   starts  = (const int*)  d_in[1];
    const int*   lengths = (const int*)  d_in[2];
    const int*   hts     = (const int*)  d_in[3];
    const float* thr     = (const float*)d_in[4];
    const float* W1      = (const float*)d_in[5];
    const float* b1      = (const float*)d_in[6];
    const float* W2      = (const float*)d_in[7];
    const float* b2      = (const float*)d_in[8];

    float* ws     = (float*)d_ws;
    float* ehat   = ws + EHAT_OFF;
    float* sim    = ws + SIM_OFF;
    float* part   = ws + PART_OFF;
    float* stdArr = ws + STD_OFF;

    pool_norm_kernel<<<Bc * NEc, 256, 0, stream>>>(x, starts, lengths, ehat);
    sim_wmma_kernel<<<Bc * NTILE, 32, 0, stream>>>(ehat, sim, part);
    std_kernel<<<Bc, 64, 0, stream>>>(part, stdArr);

    const int total = Bc * NPAIR;
    pair_mlp_kernel<<<(total + 255) / 256, 256, 0, stream>>>(
        sim, stdArr, hts, thr, W1, b1, W2, b2, (float*)d_out);
}